// GCN_31250182046298
// MI455X (gfx1250) — compile-verified
//
#include <hip/hip_runtime.h>
#include <hip/hip_bf16.h>
#include <math.h>

// ---------------------------------------------------------------------------
// GCN forward on MI455X (gfx1250, wave32):
//   h  = spmm(adj, dropout(x) @ W1)            [N,16]
//   probs = -mean_i max_k GMM-loglik(h_i)
//   h2 = spmm(adj, dropout(relu(h)) @ W2)      [N,16]
//   out = log_softmax(h2) , probs
//
// GEMMs use V_WMMA_F32_16X16X4_F32 (fp32 precision per reference).
// GEMM1 is the HBM-bound hot spot (x = 205MB); everything else is L2-resident.
// ---------------------------------------------------------------------------

typedef float v2f __attribute__((ext_vector_type(2)));
typedef float v8f __attribute__((ext_vector_type(8)));

#define F_IN 512
#define DCH  16     // hidden / class / mixture dims are all 16

// Deterministic per-element dropout, keep = 0.5, scale = 2.0.
// 4 VALU ops: add(idx precombined with salt) is folded, mul_lo, ashr, and.
// No compare/cndmask: sign bit of the hash selects keep, applied as a bitmask.
__device__ __forceinline__ float dropout_apply(float v, unsigned idx_plus_salt) {
    const unsigned h = idx_plus_salt * 2654435761u;
    const unsigned m = (unsigned)((int)h >> 31);      // all-ones iff keep
    const float    d = v + v;                         // 1/keep = 2.0 rescale
    return __uint_as_float(m & __float_as_uint(d));
}

// ---------------------------------------------------------------------------
// Kernel 1: fused dropout(x) @ W1 -> support1 [N,16]
// One 16x16 output tile per wave; K-loop over 512 in steps of 4 (WMMA f32).
// W1 (512x16 = 32KB) staged in LDS once per block (coalesced), frags via ds.
// ---------------------------------------------------------------------------
__global__ void __launch_bounds__(256)
gemm1_dropout_wmma(const float* __restrict__ x, const float* __restrict__ W1,
                   float* __restrict__ out, int N) {
    __shared__ float sW[F_IN * DCH];
    for (int i = threadIdx.x; i < (F_IN * DCH) / 4; i += 256)
        ((float4*)sW)[i] = ((const float4*)W1)[i];
    __syncthreads();

    const int lane = threadIdx.x & 31;
    const int wave = threadIdx.x >> 5;
    const int tile = blockIdx.x * 8 + wave;
    const int m0   = tile * 16;
    if (m0 >= N) return;                    // whole wave exits together

    const int  mloc  = lane & 15;           // A row / B col within tile
    const int  kh    = (lane >> 4) << 1;    // K sub-offset: 0 or 2
    const int  mrow  = min(m0 + mloc, N - 1);
    const float* xrow = x + (size_t)mrow * F_IN;
    const unsigned isalt = (unsigned)mrow * (unsigned)F_IN + 0x9E3779B9u;

    v8f acc = {};
    for (int k0 = 0; k0 < F_IN; k0 += 32) {
        // speculative stream prefetch ~512B ahead (global_prefetch_b8)
        __builtin_prefetch(xrow + k0 + 128, 0, 3);
#pragma unroll
        for (int kk = 0; kk < 32; kk += 4) {
            const int k = k0 + kk + kh;
            v2f a, b;
            a[0] = dropout_apply(xrow[k],     isalt + (unsigned)k);
            a[1] = dropout_apply(xrow[k + 1], isalt + (unsigned)k + 1);
            b[0] = sW[k * DCH + mloc];
            b[1] = sW[(k + 1) * DCH + mloc];
            acc = __builtin_amdgcn_wmma_f32_16x16x4_f32(
                false, a, false, b, (short)0, acc, false, false);
        }
    }

    const int mstore = m0 + ((lane >> 4) << 3);   // M = r + 8*(lane>=16)
    if (m0 + 16 <= N) {                           // wave-uniform fast path
#pragma unroll
        for (int r = 0; r < 8; ++r)
            out[(size_t)(mstore + r) * DCH + mloc] = acc[r];
    } else {
#pragma unroll
        for (int r = 0; r < 8; ++r) {
            const int mm = mstore + r;
            if (mm < N) out[(size_t)mm * DCH + mloc] = acc[r];
        }
    }
}

// ---------------------------------------------------------------------------
// Kernel 2: SPMM. row[] is sorted; per-node edge range found via binary
// search (redundant across the 16-lane group -> no divergence).
// lane = feature channel. Edge metadata (vals/col) is loaded coalesced 16
// edges at a time and broadcast within the half-wave via __shfl; the gather
// of support[col[e]] is one coalesced 64B line per edge, L2-resident (6.4MB).
// ---------------------------------------------------------------------------
__global__ void __launch_bounds__(256)
spmm_kernel(const float* __restrict__ vals, const int* __restrict__ row,
            const int* __restrict__ col, const float* __restrict__ sup,
            float* __restrict__ out, int N, int E) {
    const int gid    = blockIdx.x * 256 + threadIdx.x;
    const int node   = gid >> 4;
    const int f      = threadIdx.x & 15;
    const int base16 = threadIdx.x & 16;    // half-wave base lane for shuffles
    if (node >= N) return;

    int lo = 0, hi = E;
    while (lo < hi) { int mid = (lo + hi) >> 1; if (row[mid] <  node) lo = mid + 1; else hi = mid; }
    const int e_beg = lo;
    hi = E;
    while (lo < hi) { int mid = (lo + hi) >> 1; if (row[mid] <= node) lo = mid + 1; else hi = mid; }
    const int e_end = lo;

    float acc = 0.0f;
    int e = e_beg;
    for (; e + 16 <= e_end; e += 16) {      // coalesced 16-edge chunks
        const float v16 = vals[e + f];
        const int   c16 = col[e + f];
#pragma unroll
        for (int j = 0; j < 16; ++j) {
            const float v = __shfl(v16, base16 + j);
            const int   c = __shfl(c16, base16 + j);
            acc += v * sup[(size_t)c * DCH + f];
        }
    }
    for (; e < e_end; ++e)                  // remainder edges
        acc += vals[e] * sup[(size_t)col[e] * DCH + f];

    out[(size_t)node * DCH + f] = acc;
}

// ---------------------------------------------------------------------------
// Kernel 3a: per-component GMM constants kc[k] = log PI_k + 0.5*sum log PRE
//            - 8 * log(2*pi)
// ---------------------------------------------------------------------------
__global__ void kconst_kernel(const float* __restrict__ PI,
                              const float* __restrict__ PREs,
                              float* __restrict__ kc) {
    const int k = threadIdx.x;
    if (k < DCH) {
        float s = logf(PI[k]);
#pragma unroll
        for (int d = 0; d < DCH; ++d) s += 0.5f * logf(PREs[k * DCH + d]);
        kc[k] = s - 8.0f * logf(6.283185307179586f);
    }
}

__global__ void zero_scalar_kernel(float* __restrict__ p) { p[0] = 0.0f; }

// ---------------------------------------------------------------------------
// Kernel 3b: probs = -mean_i max_k logcomp[i,k]; block tree-reduce + one
//            global_atomic_add_f32 per block.
// ---------------------------------------------------------------------------
__global__ void __launch_bounds__(256)
gmm_kernel(const float* __restrict__ h, const float* __restrict__ MUs,
           const float* __restrict__ PREs, const float* __restrict__ kc,
           float* __restrict__ scalar_out, int N) {
    __shared__ float sMU[DCH * DCH], sPRE[DCH * DCH], sKC[DCH], red[256];
    if (threadIdx.x < DCH * DCH) {
        sMU[threadIdx.x]  = MUs[threadIdx.x];
        sPRE[threadIdx.x] = PREs[threadIdx.x];
    }
    if (threadIdx.x < DCH) sKC[threadIdx.x] = kc[threadIdx.x];
    __syncthreads();

    const int i = blockIdx.x * 256 + threadIdx.x;
    float best = 0.0f;
    if (i < N) {
        float hv[DCH];
        const float4* hp = (const float4*)(h + (size_t)i * DCH);
#pragma unroll
        for (int q = 0; q < 4; ++q) {
            float4 t = hp[q];
            hv[4 * q] = t.x; hv[4 * q + 1] = t.y; hv[4 * q + 2] = t.z; hv[4 * q + 3] = t.w;
        }
        best = -3.4e38f;
        for (int k = 0; k < DCH; ++k) {
            float s = sKC[k];
#pragma unroll
            for (int d = 0; d < DCH; ++d) {
                const float df = hv[d] - sMU[k * DCH + d];
                s -= 0.5f * df * df * sPRE[k * DCH + d];
            }
            best = fmaxf(best, s);
        }
    }
    red[threadIdx.x] = best;
    __syncthreads();
    for (int s2 = 128; s2 > 0; s2 >>= 1) {
        if (threadIdx.x < s2) red[threadIdx.x] += red[threadIdx.x + s2];
        __syncthreads();
    }
    if (threadIdx.x == 0) atomicAdd(scalar_out, -red[0] / (float)N);
}

// ---------------------------------------------------------------------------
// Kernel 4: fused relu + dropout + (. @ W2) via WMMA, K = 16 (4 steps).
// ---------------------------------------------------------------------------
__global__ void __launch_bounds__(256)
gemm2_relu_dropout_wmma(const float* __restrict__ h, const float* __restrict__ W2,
                        float* __restrict__ out, int N) {
    __shared__ float sW[DCH * DCH];
    if (threadIdx.x < (DCH * DCH) / 4)
        ((float4*)sW)[threadIdx.x] = ((const float4*)W2)[threadIdx.x];
    __syncthreads();

    const int lane = threadIdx.x & 31;
    const int wave = threadIdx.x >> 5;
    const int tile = blockIdx.x * 8 + wave;
    const int m0   = tile * 16;
    if (m0 >= N) return;

    const int  mloc  = lane & 15;
    const int  kh    = (lane >> 4) << 1;
    const int  mrow  = min(m0 + mloc, N - 1);
    const float* hrow = h + (size_t)mrow * DCH;
    const unsigned isalt = (unsigned)mrow * (unsigned)DCH + 0x7F4A7C15u;

    v8f acc = {};
#pragma unroll
    for (int k0 = 0; k0 < DCH; k0 += 4) {
        const int k = k0 + kh;
        v2f a, b;
        a[0] = dropout_apply(fmaxf(hrow[k],     0.0f), isalt + (unsigned)k);
        a[1] = dropout_apply(fmaxf(hrow[k + 1], 0.0f), isalt + (unsigned)k + 1);
        b[0] = sW[k * DCH + mloc];
        b[1] = sW[(k + 1) * DCH + mloc];
        acc = __builtin_amdgcn_wmma_f32_16x16x4_f32(
            false, a, false, b, (short)0, acc, false, false);
    }

    const int mstore = m0 + ((lane >> 4) << 3);
    if (m0 + 16 <= N) {                           // wave-uniform fast path
#pragma unroll
        for (int r = 0; r < 8; ++r)
            out[(size_t)(mstore + r) * DCH + mloc] = acc[r];
    } else {
#pragma unroll
        for (int r = 0; r < 8; ++r) {
            const int mm = mstore + r;
            if (mm < N) out[(size_t)mm * DCH + mloc] = acc[r];
        }
    }
}

// ---------------------------------------------------------------------------
// Kernel 5: row-wise log_softmax over 16 classes.
// ---------------------------------------------------------------------------
__global__ void __launch_bounds__(256)
logsoftmax_kernel(const float* __restrict__ h2, float* __restrict__ out, int N) {
    const int i = blockIdx.x * 256 + threadIdx.x;
    if (i >= N) return;
    float v[DCH];
    const float4* hp = (const float4*)(h2 + (size_t)i * DCH);
#pragma unroll
    for (int q = 0; q < 4; ++q) {
        float4 t = hp[q];
        v[4 * q] = t.x; v[4 * q + 1] = t.y; v[4 * q + 2] = t.z; v[4 * q + 3] = t.w;
    }
    float m = v[0];
#pragma unroll
    for (int j = 1; j < DCH; ++j) m = fmaxf(m, v[j]);
    float se = 0.0f;
#pragma unroll
    for (int j = 0; j < DCH; ++j) se += expf(v[j] - m);
    const float lse = m + logf(se);
    float4* op = (float4*)(out + (size_t)i * DCH);
#pragma unroll
    for (int q = 0; q < 4; ++q) {
        float4 t;
        t.x = v[4 * q]     - lse;
        t.y = v[4 * q + 1] - lse;
        t.z = v[4 * q + 2] - lse;
        t.w = v[4 * q + 3] - lse;
        op[q] = t;
    }
}

// ---------------------------------------------------------------------------
extern "C" void kernel_launch(void* const* d_in, const int* in_sizes, int n_in,
                              void* d_out, int out_size, void* d_ws, size_t ws_size,
                              hipStream_t stream) {
    const float* x    = (const float*)d_in[0];
    const float* W1   = (const float*)d_in[1];
    const float* W2   = (const float*)d_in[2];
    const float* adj  = (const float*)d_in[3];
    const float* PI   = (const float*)d_in[4];
    const float* MUs  = (const float*)d_in[5];
    const float* PREs = (const float*)d_in[6];
    const int*   row  = (const int*)d_in[7];
    const int*   col  = (const int*)d_in[8];

    const int N = in_sizes[0] / F_IN;
    const int E = in_sizes[3];

    float* out        = (float*)d_out;
    float* scalar_out = out + (size_t)N * DCH;          // probs slot

    float* bufA = (float*)d_ws;                         // support1, later h2
    float* bufB = bufA + (size_t)N * DCH;               // h
    float* bufC = bufB + (size_t)N * DCH;               // support2
    float* kc   = bufC + (size_t)N * DCH;               // 16 GMM constants

    const int tiles  = (N + 15) / 16;
    const int gGemm  = (tiles + 7) / 8;                 // 8 waves (tiles) / block
    const int gSpmm  = (N * DCH + 255) / 256;
    const int gNode  = (N + 255) / 256;

    zero_scalar_kernel<<<1, 1, 0, stream>>>(scalar_out);
    kconst_kernel<<<1, 32, 0, stream>>>(PI, PREs, kc);

    gemm1_dropout_wmma<<<gGemm, 256, 0, stream>>>(x, W1, bufA, N);
    spmm_kernel<<<gSpmm, 256, 0, stream>>>(adj, row, col, bufA, bufB, N, E);
    gmm_kernel<<<gNode, 256, 0, stream>>>(bufB, MUs, PREs, kc, scalar_out, N);
    gemm2_relu_dropout_wmma<<<gGemm, 256, 0, stream>>>(bufB, W2, bufC, N);
    spmm_kernel<<<gSpmm, 256, 0, stream>>>(adj, row, col, bufC, bufA, N, E);
    logsoftmax_kernel<<<gNode, 256, 0, stream>>>(bufA, out, N);
}